// MultiTableFit_15719580304098
// MI455X (gfx1250) — compile-verified
//
#include <hip/hip_runtime.h>
#include <cstdint>

// Elementwise piecewise-LUT sigmoid approximation.
// Memory-bound: 8 B/element * 33.5M elements = 268 MB @ 23.3 TB/s -> ~11.5 us floor.
// No matmul structure -> WMMA not applicable; optimize the data-movement path:
//  - async global->LDS staging of the two 256-entry LUTs (gfx1250 ASYNCcnt path)
//  - single ds_load_b32 gather per element (dense/sparse regions are exclusive)
//  - B128 non-temporal loads/stores for the 268 MB stream (don't thrash L2)

#define TABLE_SIZE 256

typedef int   v4i __attribute__((ext_vector_type(4)));
typedef float v4f __attribute__((ext_vector_type(4)));

__device__ __forceinline__ float eval1(int code, float data_scale,
                                       const float* __restrict__ lut,
                                       float ll_ymin, float ll_slope,
                                       float rl_ymin, float rl_slope,
                                       float left_c, float right_c)
{
    // dequantize: codes in [0,65536), zero point 32768
    float x = (float)(code - 32768) * data_scale;

    // dense bins over [-3,3]: idx = clip(round((x+3) * 255/6), 0, 255)
    float df = __builtin_rintf((x + 3.0f) * (255.0f / 6.0f));
    df = __builtin_fminf(__builtin_fmaxf(df, 0.0f), 255.0f);
    // sparse bins over [-6,6]: idx = clip(round((x+6) * 255/12), 0, 255)
    float sf = __builtin_rintf((x + 6.0f) * (255.0f / 12.0f));
    sf = __builtin_fminf(__builtin_fmaxf(sf, 0.0f), 255.0f);

    // dense and sparse regions are mutually exclusive -> ONE LDS gather
    bool in_dense = (x >= -3.0f) && (x <= 3.0f);
    int idx = in_dense ? (int)df : ((int)sf + TABLE_SIZE);
    float tval = lut[idx];                       // ds_load_b32

    // shoulder line fits (slopes precomputed: (ymax-ymin)/4)
    float ll = __builtin_fmaf(x + 10.0f, ll_slope, ll_ymin);
    float rl = __builtin_fmaf(x - 6.0f,  rl_slope, rl_ymin);

    float y = tval;
    y = (x < -6.0f) ? ((x < -10.0f) ? left_c  : ll) : y;
    y = (x >  6.0f) ? ((x >  10.0f) ? right_c : rl) : y;
    return y;
}

__global__ __launch_bounds__(256) void MultiTableFit_kernel(
    const int*   __restrict__ data,
    const float* __restrict__ data_scale_p,
    const float* __restrict__ dense_table,
    const float* __restrict__ sparse_table,
    const float* __restrict__ ll_ymin_p,  const float* __restrict__ ll_ymax_p,
    const float* __restrict__ rl_ymin_p,  const float* __restrict__ rl_ymax_p,
    const float* __restrict__ left_c_p,   const float* __restrict__ right_c_p,
    float* __restrict__ out, int nvec, int n)
{
    __shared__ float lut[2 * TABLE_SIZE];   // [0..255] dense, [256..511] sparse

    const int t = threadIdx.x;              // block = 256 threads = 8 wave32

    // ---- stage both LUTs into LDS via gfx1250 async global->LDS DMA ----
    {
        uint32_t lds0 = (uint32_t)(uintptr_t)(&lut[t]);              // low 32 bits of
        uint32_t lds1 = (uint32_t)(uintptr_t)(&lut[TABLE_SIZE + t]); // flat ptr = LDS offset
        uint32_t voff = (uint32_t)t * 4u;
        asm volatile(
            "global_load_async_to_lds_b32 %0, %2, %3\n\t"
            "global_load_async_to_lds_b32 %1, %2, %4\n\t"
            "s_wait_asynccnt 0"
            :
            : "v"(lds0), "v"(lds1), "v"(voff),
              "s"(dense_table), "s"(sparse_table)
            : "memory");
    }
    __syncthreads();   // s_barrier_signal/s_barrier_wait on gfx1250

    // ---- uniform scalar params (s_load path) ----
    const float data_scale = *data_scale_p;
    const float ll_ymin  = *ll_ymin_p;
    const float ll_slope = (*ll_ymax_p - ll_ymin) * 0.25f;  // /(LL_XMAX-LL_XMIN)=4
    const float rl_ymin  = *rl_ymin_p;
    const float rl_slope = (*rl_ymax_p - rl_ymin) * 0.25f;  // /(RL_XMAX-RL_XMIN)=4
    const float left_c   = *left_c_p;
    const float right_c  = *right_c_p;

    // ---- streaming main loop: B128 NT loads/stores, 4 elements per thread/iter ----
    const v4i* __restrict__ in4  = reinterpret_cast<const v4i*>(data);
    v4f*       __restrict__ out4 = reinterpret_cast<v4f*>(out);
    const int gid    = blockIdx.x * blockDim.x + t;
    const int stride = gridDim.x * blockDim.x;

    for (int i = gid; i < nvec; i += stride) {
        v4i c = __builtin_nontemporal_load(&in4[i]);
        v4f r;
        r.x = eval1(c.x, data_scale, lut, ll_ymin, ll_slope, rl_ymin, rl_slope, left_c, right_c);
        r.y = eval1(c.y, data_scale, lut, ll_ymin, ll_slope, rl_ymin, rl_slope, left_c, right_c);
        r.z = eval1(c.z, data_scale, lut, ll_ymin, ll_slope, rl_ymin, rl_slope, left_c, right_c);
        r.w = eval1(c.w, data_scale, lut, ll_ymin, ll_slope, rl_ymin, rl_slope, left_c, right_c);
        __builtin_nontemporal_store(r, &out4[i]);
    }

    // ---- scalar tail (n is a multiple of 4 in practice; kept for safety) ----
    const int rem  = n & 3;
    const int base = nvec << 2;
    if (gid < rem) {
        out[base + gid] = eval1(data[base + gid], data_scale, lut,
                                ll_ymin, ll_slope, rl_ymin, rl_slope, left_c, right_c);
    }
}

extern "C" void kernel_launch(void* const* d_in, const int* in_sizes, int n_in,
                              void* d_out, int out_size, void* d_ws, size_t ws_size,
                              hipStream_t stream) {
    // setup_inputs() order:
    // 0 data(int32), 1 data_scale, 2 scale(unused), 3 dense_table, 4 sparse_table,
    // 5 ll_ymin, 6 ll_ymax, 7 rl_ymin, 8 rl_ymax, 9 left_const_y, 10 right_const_y
    const int*   data         = (const int*)  d_in[0];
    const float* data_scale   = (const float*)d_in[1];
    const float* dense_table  = (const float*)d_in[3];
    const float* sparse_table = (const float*)d_in[4];
    const float* ll_ymin      = (const float*)d_in[5];
    const float* ll_ymax      = (const float*)d_in[6];
    const float* rl_ymin      = (const float*)d_in[7];
    const float* rl_ymax      = (const float*)d_in[8];
    const float* left_c       = (const float*)d_in[9];
    const float* right_c      = (const float*)d_in[10];
    float* out = (float*)d_out;

    const int n    = in_sizes[0];        // 33,554,432
    const int nvec = n >> 2;             // int4 chunks

    const int block = 256;               // 8 wave32 per block
    int grid = 8192;                     // ~2M threads -> 4 iters/thread
    int maxg = (nvec + block - 1) / block;
    if (grid > maxg) grid = maxg;
    if (grid < 1) grid = 1;

    MultiTableFit_kernel<<<grid, block, 0, stream>>>(
        data, data_scale, dense_table, sparse_table,
        ll_ymin, ll_ymax, rl_ymin, rl_ymax, left_c, right_c,
        out, nvec, n);
}